// LELoss_54022098649765
// MI455X (gfx1250) — compile-verified
//
#include <hip/hip_runtime.h>

typedef float v2f __attribute__((ext_vector_type(2)));
typedef float v4f __attribute__((ext_vector_type(4)));
typedef float v8f __attribute__((ext_vector_type(8)));

#define B_ROWS 8192
#define D_DIM  1024
#define E_DIM  128
#define I_DIM  20   // intrinsic

// ---------------------------------------------------------------------------
// ws layout (floats): ws[0]=recon sum, ws[1]=pca sum, ws[2]=proj sum
// ---------------------------------------------------------------------------
__global__ void zero_ws_kernel(float* ws) {
    if (threadIdx.x < 4) ws[threadIdx.x] = 0.0f;
}

// ---------------------------------------------------------------------------
// Sum (x - decoded)^2 over 8M elements. Memory bound: float4 non-temporal
// streams, wave32 shuffle reduce, one atomic per wave.
// ---------------------------------------------------------------------------
__global__ __launch_bounds__(256) void recon_kernel(
    const float* __restrict__ x, const float* __restrict__ dec,
    float* __restrict__ acc) {
    const int n4 = (B_ROWS * D_DIM) / 4;
    const int stride = gridDim.x * blockDim.x;
    float s = 0.0f;
    for (int i = blockIdx.x * blockDim.x + threadIdx.x; i < n4; i += stride) {
        v4f a = __builtin_nontemporal_load(((const v4f*)x)   + i);
        v4f b = __builtin_nontemporal_load(((const v4f*)dec) + i);
        v4f d = a - b;
        s += d.x * d.x + d.y * d.y + d.z * d.z + d.w * d.w;
    }
    #pragma unroll
    for (int off = 16; off > 0; off >>= 1) s += __shfl_down(s, off, 32);
    if ((threadIdx.x & 31) == 0) atomicAdd(acc, s);
}

// ---------------------------------------------------------------------------
// PCA term: z = latent @ rsrA^T computed as 16x16 tiles via
// V_WMMA_F32_16X16X4_F32 (K=4, 5 steps for I=20), fused with
// sum((encoded - z)^2). One tile per wave, 8 waves per block.
//
// A fragment (16x4 f32): lane L -> row M = L%16; vgpr v -> K = v + 2*(L/16)
// B fragment (4x16 f32): lane L -> col N = L%16; vgpr v -> K = v + 2*(L/16)
//   B(k,n) = rsrA^T[k][n] = rsrA[n][k]   (rsrA is [E, I] row-major)
// C/D (16x16 f32): vgpr r -> row M = r + 8*(L/16); lane -> col N = L%16
// ---------------------------------------------------------------------------
__global__ __launch_bounds__(256) void pca_wmma_kernel(
    const float* __restrict__ latent, const float* __restrict__ rsrA,
    const float* __restrict__ encoded, float* __restrict__ acc) {
    const int lane = threadIdx.x & 31;
    const int wave = threadIdx.x >> 5;
    const int tilesN = E_DIM / 16;                 // 8
    const int tile = blockIdx.x * 8 + wave;        // 4096 tiles total
    const int m0 = (tile / tilesN) * 16;
    const int n0 = (tile % tilesN) * 16;
    const int half = lane >> 4;                    // 0 or 1
    const int l15  = lane & 15;

    const float* arow = latent + (size_t)(m0 + l15) * I_DIM + half * 2;
    const float* brow = rsrA   + (size_t)(n0 + l15) * I_DIM + half * 2;

    v8f c = {};
    #pragma unroll
    for (int k0 = 0; k0 < I_DIM; k0 += 4) {        // 5 WMMAs, chained on C
        v2f a = *(const v2f*)(arow + k0);
        v2f b = *(const v2f*)(brow + k0);
        c = __builtin_amdgcn_wmma_f32_16x16x4_f32(
                /*neg_a=*/false, a, /*neg_b=*/false, b,
                /*c_mod=*/(short)0, c, /*reuse_a=*/false, /*reuse_b=*/false);
    }

    float s = 0.0f;
    #pragma unroll
    for (int r = 0; r < 8; ++r) {
        const int m = m0 + r + 8 * half;
        const float e = encoded[(size_t)m * E_DIM + n0 + l15];
        const float d = c[r] - e;
        s += d * d;
    }
    #pragma unroll
    for (int off = 16; off > 0; off >>= 1) s += __shfl_down(s, off, 32);
    if (lane == 0) atomicAdd(acc, s);
}

// ---------------------------------------------------------------------------
// Proj term: sum((rsrA^T rsrA - I)^2) over 20x20=400 gram entries, each a
// dot over E=128. Tiny: one block, one entry per thread.
// ---------------------------------------------------------------------------
__global__ __launch_bounds__(512) void proj_kernel(
    const float* __restrict__ rsrA, float* __restrict__ ws) {
    __shared__ float red[512];
    const int t = threadIdx.x;
    float v = 0.0f;
    if (t < I_DIM * I_DIM) {
        const int i = t / I_DIM, j = t % I_DIM;
        float d = 0.0f;
        #pragma unroll 4
        for (int e = 0; e < E_DIM; ++e)
            d += rsrA[e * I_DIM + i] * rsrA[e * I_DIM + j];
        if (i == j) d -= 1.0f;
        v = d * d;
    }
    red[t] = v;
    __syncthreads();
    #pragma unroll
    for (int sft = 256; sft > 0; sft >>= 1) {
        if (t < sft) red[t] += red[t + sft];
        __syncthreads();
    }
    if (t == 0) ws[2] = red[0];
}

// ---------------------------------------------------------------------------
// Combine: loss = ws0/B + (1.0 + 0.1) * ws1/B + 0.1 * ws2/400
// ---------------------------------------------------------------------------
__global__ void finalize_kernel(const float* __restrict__ ws,
                                float* __restrict__ out) {
    if (threadIdx.x == 0 && blockIdx.x == 0) {
        const float invB = 1.0f / (float)B_ROWS;
        out[0] = ws[0] * invB
               + 1.1f * ws[1] * invB
               + 0.1f * ws[2] / (float)(I_DIM * I_DIM);
    }
}

extern "C" void kernel_launch(void* const* d_in, const int* in_sizes, int n_in,
                              void* d_out, int out_size, void* d_ws, size_t ws_size,
                              hipStream_t stream) {
    const float* x       = (const float*)d_in[0];  // [8192,1024]
    const float* encoded = (const float*)d_in[1];  // [8192,128]
    const float* latent  = (const float*)d_in[2];  // [8192,20]
    const float* decoded = (const float*)d_in[3];  // [8192,1024]
    const float* rsrA    = (const float*)d_in[4];  // [128,20]
    float* out = (float*)d_out;
    float* ws  = (float*)d_ws;

    zero_ws_kernel<<<1, 32, 0, stream>>>(ws);

    // 64 MB stream: plenty of blocks to saturate 23.3 TB/s
    recon_kernel<<<2048, 256, 0, stream>>>(x, decoded, ws + 0);

    // 4096 tiles of 16x16, 8 waves (tiles) per block -> 512 blocks
    pca_wmma_kernel<<<512, 256, 0, stream>>>(latent, rsrA, encoded, ws + 1);

    proj_kernel<<<1, 512, 0, stream>>>(rsrA, ws);

    finalize_kernel<<<1, 32, 0, stream>>>(ws, out);
}